// HybridRegressionModel_65481071396201
// MI455X (gfx1250) — compile-verified
//
#include <hip/hip_runtime.h>

typedef __attribute__((ext_vector_type(2))) float v2f;
typedef __attribute__((ext_vector_type(4))) float v4f;
typedef __attribute__((ext_vector_type(8))) float v8f;

#define NF   256   // NUM_FEATURES
#define RSZ  64    // REF_SIZE
#define WAVES 4    // waves per block
#define RPW  32    // rows per wave (two 16x16 WMMA M-tiles)

// padded LDS row strides (floats) to break bank conflicts
#define CR_S 65    // cross rows
#define H1_S 33    // h1 rows
#define H2_S 17    // h2 rows

// D = A(16x4) x B(4x16) + C, f32 WMMA.
// A layout: lanes 0-15 -> M=lane, v0=K0,v1=K1 ; lanes 16-31 -> M=lane-16, v0=K2,v1=K3
// B layout: lanes 0-15 -> N=lane, v0=K0,v1=K1 ; lanes 16-31 -> N=lane-16, v0=K2,v1=K3
// C/D layout: vgpr v, lanes 0-15 -> (M=v, N=lane); lanes 16-31 -> (M=v+8, N=lane-16)
__device__ __forceinline__ v8f wmma_f32(v2f a, v2f b, v8f c) {
  return __builtin_amdgcn_wmma_f32_16x16x4_f32(
      /*neg_a=*/false, a, /*neg_b=*/false, b,
      /*c_mod=*/(short)0, c, /*reuse_a=*/false, /*reuse_b=*/false);
}

__global__ __launch_bounds__(WAVES * 32) void
HybridRegressionModel_65481071396201_kernel(
    const float* __restrict__ x,  const float* __restrict__ W1,
    const float* __restrict__ b1, const float* __restrict__ W2,
    const float* __restrict__ b2, const float* __restrict__ ref,
    const float* __restrict__ Wf, const float* __restrict__ bf,
    float* __restrict__ out, int B)
{
  __shared__ float lds_r2[RSZ];
  __shared__ float lds_cross[WAVES][RPW * CR_S];  // ~8.3 KB / wave
  __shared__ float lds_h1[WAVES][RPW * H1_S];     // ~4.2 KB / wave
  __shared__ float lds_h2[WAVES][RPW * H2_S];     // ~2.2 KB / wave
  __shared__ float lds_x2[WAVES][2][32];

  const int tid  = threadIdx.x;
  const int wave = tid >> 5;
  const int lane = tid & 31;
  const int hi   = lane >> 4;   // upper half of wave
  const int lo   = lane & 15;

  // --- per-block: ||r_j||^2 into LDS (weights are L2/WGP$ resident) ---
  if (tid < RSZ) {
    const v4f* rp = (const v4f*)(ref + (size_t)tid * NF);
    float s = 0.f;
    #pragma unroll 8
    for (int i = 0; i < NF / 4; ++i) {
      v4f r = rp[i];
      s += r.x * r.x + r.y * r.y + r.z * r.z + r.w * r.w;
    }
    lds_r2[tid] = s;
  }

  const int row0 = (blockIdx.x * WAVES + wave) * RPW;

  // --- main K loop: 12 WMMAs / step (2 M-tiles share 6 B-fragments) ---
  v8f accC[2][4] = {};          // cross accumulators: [mtile][ntile]
  v8f accH[2][2] = {};          // h1 accumulators:    [mtile][ntile]
  float sxA = 0.f, sxB = 0.f;   // partial ||x_m||^2 per M-tile
  const float* xrowA = x + (size_t)(row0 + lo) * NF;
  const float* xrowB = x + (size_t)(row0 + 16 + lo) * NF;

  #pragma unroll 2
  for (int k = 0; k < NF; k += 4) {
    const int kk = k + 2 * hi;
    v2f aA = *(const v2f*)(xrowA + kk);    // global_load_b64, coalesced stream
    v2f aB = *(const v2f*)(xrowB + kk);
    sxA += aA.x * aA.x + aA.y * aA.y;
    sxB += aB.x * aB.x + aB.y * aB.y;

    v2f br[4], bw[2];
    #pragma unroll
    for (int t = 0; t < 4; ++t)
      br[t] = *(const v2f*)(ref + (size_t)(lo + 16 * t) * NF + kk);
    #pragma unroll
    for (int t = 0; t < 2; ++t) {
      bw[t].x = W1[kk * 32 + lo + 16 * t];
      bw[t].y = W1[(kk + 1) * 32 + lo + 16 * t];
    }

    #pragma unroll
    for (int t = 0; t < 4; ++t) {
      accC[0][t] = wmma_f32(aA, br[t], accC[0][t]);
      accC[1][t] = wmma_f32(aB, br[t], accC[1][t]);
    }
    #pragma unroll
    for (int t = 0; t < 2; ++t) {
      accH[0][t] = wmma_f32(aA, bw[t], accH[0][t]);
      accH[1][t] = wmma_f32(aB, bw[t], accH[1][t]);
    }
  }

  // --- spill cross + relu(h1 + b1) + x2 partials to LDS ---
  #pragma unroll
  for (int mt = 0; mt < 2; ++mt) {
    #pragma unroll
    for (int v = 0; v < 8; ++v) {
      const int m = mt * 16 + v + 8 * hi;
      #pragma unroll
      for (int t = 0; t < 4; ++t)
        lds_cross[wave][m * CR_S + lo + 16 * t] = accC[mt][t][v];
      lds_h1[wave][m * H1_S + lo]      = fmaxf(accH[mt][0][v] + b1[lo],      0.f);
      lds_h1[wave][m * H1_S + lo + 16] = fmaxf(accH[mt][1][v] + b1[lo + 16], 0.f);
    }
  }
  lds_x2[wave][0][lane] = sxA;  // x2[m] = [mt][m&15] + [mt][(m&15)+16]
  lds_x2[wave][1][lane] = sxB;
  __syncthreads();

  // --- h2 = relu(h1 @ W2 + b2): two 16x16 tiles, K=32 -> 16 WMMAs ---
  #pragma unroll
  for (int mt = 0; mt < 2; ++mt) {
    v8f accO = {};
    #pragma unroll
    for (int k = 0; k < 32; k += 4) {
      const int kk = k + 2 * hi;
      v2f a, b;
      a.x = lds_h1[wave][(mt * 16 + lo) * H1_S + kk];
      a.y = lds_h1[wave][(mt * 16 + lo) * H1_S + kk + 1];
      b.x = W2[kk * 16 + lo];
      b.y = W2[(kk + 1) * 16 + lo];
      accO = wmma_f32(a, b, accO);
    }
    #pragma unroll
    for (int v = 0; v < 8; ++v) {
      const int m = mt * 16 + v + 8 * hi;
      lds_h2[wave][m * H2_S + lo] = fmaxf(accO[v] + b2[lo], 0.f);
    }
  }
  __syncthreads();

  // --- epilogue: all 32 lanes, one output row each ---
  {
    const int m    = lane;
    const int mt   = m >> 4;
    const int rr   = m & 15;
    const int grow = row0 + m;
    const float x2v = lds_x2[wave][mt][rr] + lds_x2[wave][mt][rr + 16];
    float sum = bf[0];
    #pragma unroll 4
    for (int n = 0; n < 16; ++n)
      sum += lds_h2[wave][m * H2_S + n] * Wf[n];
    #pragma unroll 4
    for (int j = 0; j < RSZ; ++j) {
      const float d = x2v + lds_r2[j] - 2.0f * lds_cross[wave][m * CR_S + j];
      sum += __expf(-d) * Wf[16 + j];   // GAMMA = 1.0
    }
    if (grow < B) out[grow] = sum;      // 32 consecutive f32 stores, coalesced
  }
}

extern "C" void kernel_launch(void* const* d_in, const int* in_sizes, int n_in,
                              void* d_out, int out_size, void* d_ws, size_t ws_size,
                              hipStream_t stream) {
  const float* x  = (const float*)d_in[0];
  const float* W1 = (const float*)d_in[1];
  const float* b1 = (const float*)d_in[2];
  const float* W2 = (const float*)d_in[3];
  const float* b2 = (const float*)d_in[4];
  const float* rv = (const float*)d_in[5];
  const float* Wf = (const float*)d_in[6];
  const float* bf = (const float*)d_in[7];
  float* out = (float*)d_out;

  const int B = in_sizes[0] / NF;              // 131072
  const int rows_per_block = WAVES * RPW;      // 128
  const int blocks = (B + rows_per_block - 1) / rows_per_block;  // 1024

  HybridRegressionModel_65481071396201_kernel<<<blocks, WAVES * 32, 0, stream>>>(
      x, W1, b1, W2, b2, rv, Wf, bf, out, B);
}